// LTEM_29686813950260
// MI455X (gfx1250) — compile-verified
//
#include <hip/hip_runtime.h>
#include <hip/hip_bf16.h>
#include <math.h>

// ---------------------------------------------------------------------------
// Mamba-style 5-branch selective scan block for MI455X (gfx1250, wave32).
// fp32 end-to-end; GEMMs use V_WMMA_F32_16X16X4_F32 (exact fp32 matrix path).
// ---------------------------------------------------------------------------

typedef __attribute__((ext_vector_type(2))) float v2f;
typedef __attribute__((ext_vector_type(8))) float v8f;

#define NB      5
#define NBATCH  512
#define LC      200
#define DM      7
#define DS      32
#define DI      140
#define NX      65      // dt(1) + B(32) + C(32)
#define NTILE   16
#define NTILES  13      // ceil(200/16)
#define NKK     (DI/4)  // 35 wmma k-steps

__device__ __forceinline__ float sigmoidf_(float v) { return 1.f / (1.f + __expf(-v)); }
__device__ __forceinline__ float softplusf_(float v) { return (v > 20.f) ? v : log1pf(__expf(v)); }

// ===========================================================================
// K1: fused in_proj + conv + silu + x_proj(WMMA) + delta + scan + gate +
//     out_proj(WMMA).  One block per (batch, branch); 5 waves of 32.
// ===========================================================================
__global__ __launch_bounds__(160)
void k1_mamba_branch(const float* __restrict__ x,     // (512,1000,7)
                     const float* __restrict__ ipw,   // (5,280,7)
                     const float* __restrict__ cw,    // (5,140,1,3)
                     const float* __restrict__ cb,    // (5,140)
                     const float* __restrict__ xpw,   // (5,65,140)
                     const float* __restrict__ dtw,   // (5,140,1)
                     const float* __restrict__ dtb,   // (5,140)
                     const float* __restrict__ alog,  // (5,140,32)
                     const float* __restrict__ dpw,   // (5,140)
                     const float* __restrict__ opw,   // (5,7,140)
                     float* __restrict__ wsy)         // (5,512,200,7) pre-BN
{
    const int b    = blockIdx.x;
    const int br   = blockIdx.y;
    const int tid  = threadIdx.x;
    const int wave = tid >> 5;
    const int lane = tid & 31;
    const int m    = lane & 15;   // M row (A/D) or N col (B/D)
    const int hi   = lane >> 4;   // K-half selector

    __shared__ float s_xrow[18 * 8];        // staged x rows (t-2 .. t+15)
    __shared__ float s_u0  [18 * DI];       // overlay: xi_raw (A1/A2) | delta (C/D)
    __shared__ float s_xi  [NTILE * DI];    // post conv+silu
    __shared__ float s_z   [NTILE * DI];
    __shared__ float s_xdbl[NTILE * 80];    // x_proj out (N padded 65->80)
    __shared__ float s_y   [NTILE * DI];
    __shared__ float s_ipw [2 * DI * DM];   // in_proj weights (280x7)
    __shared__ float s_cw  [DI * 3];
    __shared__ float s_cb  [DI];
    __shared__ float s_dtw [DI];
    __shared__ float s_dtb [DI];
    float* const s_xraw = s_u0;             // 18*DI, live A1..A2
    float* const s_del  = s_u0;             // 16*DI, live C..D

    // Prefetch this block's x chunk (gfx1250 global_prefetch_b8 path).
    {
        const char* px = (const char*)&x[(b * 1000 + br * LC) * DM];
        for (int off = tid * 64; off < LC * DM * 4; off += (int)blockDim.x * 64)
            __builtin_prefetch(px + off, 0, 3);
    }

    // Stage loop-invariant small weights in LDS.
    for (int i = tid; i < 2 * DI * DM; i += blockDim.x) s_ipw[i] = ipw[br * 2 * DI * DM + i];
    for (int i = tid; i < DI * 3; i += blockDim.x) s_cw[i] = cw[br * DI * 3 + i];
    for (int i = tid; i < DI; i += blockDim.x) {
        s_cb[i]  = cb [br * DI + i];
        s_dtw[i] = dtw[br * DI + i];
        s_dtb[i] = dtb[br * DI + i];
    }

    // Per-thread persistent scan state: thread d (<140) owns h[d][0..31].
    float h[DS];
    float An[DS];
    float dp_d = 0.f;
    if (tid < DI) {
        dp_d = dpw[br * DI + tid];
        #pragma unroll
        for (int s = 0; s < DS; ++s) {
            An[s] = -__expf(alog[(br * DI + tid) * DS + s]);
            h[s]  = 0.f;
        }
    }

    // Branchless B-fragment bases + masks (loop invariant per lane).
    const int   eB     = wave * 16 + m;                       // x_proj out column
    const float maskB  = (eB < NX) ? 1.f : 0.f;
    const float* xpB   = &xpw[((size_t)br * NX + (eB < NX ? eB : NX - 1)) * DI];
    const float maskO  = (m < DM) ? 1.f : 0.f;
    const float* opB   = &opw[((size_t)br * DM + (m < DM ? m : DM - 1)) * DI];

    __syncthreads();

    for (int tile = 0; tile < NTILES; ++tile) {
        const int tbase = tile * NTILE;

        // ---- Phase A0: stage x rows (zero pad outside chunk) --------------
        for (int i = tid; i < 18 * DM; i += blockDim.x) {
            int r = i / DM, mm = i % DM;
            int t = tbase - 2 + r;
            s_xrow[r * 8 + mm] =
                (t >= 0 && t < LC) ? x[(b * 1000 + br * LC + t) * DM + mm] : 0.f;
        }
        __syncthreads();

        // ---- Phase A1: in_proj (K=7, VALU) for xi_raw (18 rows) and z -----
        for (int i = tid; i < 18 * DI; i += blockDim.x) {
            int r = i / DI, d = i % DI;
            const float* w = &s_ipw[d * DM];
            float acc = 0.f;
            #pragma unroll
            for (int mm = 0; mm < DM; ++mm) acc += s_xrow[r * 8 + mm] * w[mm];
            s_xraw[i] = acc;
        }
        for (int i = tid; i < NTILE * DI; i += blockDim.x) {
            int r = i / DI, d = i % DI;
            const float* w = &s_ipw[(DI + d) * DM];
            float acc = 0.f;
            #pragma unroll
            for (int mm = 0; mm < DM; ++mm) acc += s_xrow[(r + 2) * 8 + mm] * w[mm];
            s_z[i] = acc;
        }
        __syncthreads();

        // ---- Phase A2: causal depthwise conv(3) + bias + SiLU -------------
        for (int i = tid; i < NTILE * DI; i += blockDim.x) {
            int r = i / DI, d = i % DI;
            float v = s_cw[d * 3 + 0] * s_xraw[(r + 0) * DI + d]
                    + s_cw[d * 3 + 1] * s_xraw[(r + 1) * DI + d]
                    + s_cw[d * 3 + 2] * s_xraw[(r + 2) * DI + d] + s_cb[d];
            s_xi[i] = v * sigmoidf_(v);
        }
        __syncthreads();

        // ---- Phase B: x_dbl = xi(16x140) @ xpw^T(140x65) via f32 WMMA -----
        // 5 waves x one 16-wide N tile; 35 k-steps, dual accumulators.
        {
            v8f acc0 = {0.f,0.f,0.f,0.f,0.f,0.f,0.f,0.f};
            v8f acc1 = {0.f,0.f,0.f,0.f,0.f,0.f,0.f,0.f};
            const float* arow = &s_xi[m * DI];
            #pragma unroll
            for (int kk = 0; kk < NKK; kk += 2) {
                {
                    const int k = kk * 4 + hi * 2;
                    v2f a;  a[0]  = arow[k];          a[1] = arow[k + 1];
                    v2f bm; bm[0] = xpB[k] * maskB;   bm[1] = xpB[k + 1] * maskB;
                    acc0 = __builtin_amdgcn_wmma_f32_16x16x4_f32(
                               false, a, false, bm, (short)0, acc0, false, false);
                }
                if (kk + 1 < NKK) {
                    const int k = (kk + 1) * 4 + hi * 2;
                    v2f a;  a[0]  = arow[k];          a[1] = arow[k + 1];
                    v2f bm; bm[0] = xpB[k] * maskB;   bm[1] = xpB[k + 1] * maskB;
                    acc1 = __builtin_amdgcn_wmma_f32_16x16x4_f32(
                               false, a, false, bm, (short)0, acc1, false, false);
                }
            }
            #pragma unroll
            for (int r = 0; r < 8; ++r) {
                int M = hi * 8 + r;
                s_xdbl[M * 80 + eB] = acc0[r] + acc1[r];
            }
        }
        __syncthreads();

        // ---- Phase C: delta = softplus(dt * dtw + dtb) --------------------
        for (int i = tid; i < NTILE * DI; i += blockDim.x) {
            int r = i / DI, d = i % DI;
            s_del[i] = softplusf_(s_xdbl[r * 80 + 0] * s_dtw[d] + s_dtb[d]);
        }
        __syncthreads();

        // ---- Phase D: selective scan (sequential in t, reg-resident h) ----
        if (tid < DI) {
            const int d = tid;
            for (int j = 0; j < NTILE; ++j) {
                const int tg = tbase + j;
                if (tg >= LC) break;
                const float del = s_del[j * DI + d];
                const float xv  = s_xi [j * DI + d];
                float y = 0.f;
                #pragma unroll
                for (int s = 0; s < DS; ++s) {
                    float dA = __expf(del * An[s]);
                    h[s] = h[s] * dA + del * s_xdbl[j * 80 + 1 + s] * xv;
                    y += h[s] * s_xdbl[j * 80 + 33 + s];
                }
                const float zv = s_z[j * DI + d];
                s_y[j * DI + d] = (y + dp_d * xv) * (zv * sigmoidf_(zv));
            }
        }
        __syncthreads();

        // ---- Phase E: out_proj (16x140 @ 140x7) via f32 WMMA, wave 0 ------
        if (wave == 0) {
            v8f acc0 = {0.f,0.f,0.f,0.f,0.f,0.f,0.f,0.f};
            v8f acc1 = {0.f,0.f,0.f,0.f,0.f,0.f,0.f,0.f};
            const float* arow = &s_y[m * DI];
            #pragma unroll
            for (int kk = 0; kk < NKK; kk += 2) {
                {
                    const int k = kk * 4 + hi * 2;
                    v2f a;  a[0]  = arow[k];          a[1] = arow[k + 1];
                    v2f bm; bm[0] = opB[k] * maskO;   bm[1] = opB[k + 1] * maskO;
                    acc0 = __builtin_amdgcn_wmma_f32_16x16x4_f32(
                               false, a, false, bm, (short)0, acc0, false, false);
                }
                if (kk + 1 < NKK) {
                    const int k = (kk + 1) * 4 + hi * 2;
                    v2f a;  a[0]  = arow[k];          a[1] = arow[k + 1];
                    v2f bm; bm[0] = opB[k] * maskO;   bm[1] = opB[k + 1] * maskO;
                    acc1 = __builtin_amdgcn_wmma_f32_16x16x4_f32(
                               false, a, false, bm, (short)0, acc1, false, false);
                }
            }
            #pragma unroll
            for (int r = 0; r < 8; ++r) {
                const int M  = hi * 8 + r;
                const int tg = tbase + M;
                if (tg < LC && m < DM)
                    wsy[((br * NBATCH + b) * LC + tg) * DM + m] = acc0[r] + acc1[r];
            }
        }
        __syncthreads();
    }
}

// ===========================================================================
// K2: BatchNorm statistics over (batch,time) per (branch,channel).
// stats[(br*7+o)*2] = gamma/sqrt(var+eps); [..+1] = beta - mu*scale.
// ===========================================================================
__global__ __launch_bounds__(256)
void k2_bn_stats(const float* __restrict__ wsy,
                 const float* __restrict__ bn_g,
                 const float* __restrict__ bn_b,
                 float* __restrict__ stats)
{
    const int o   = blockIdx.x;   // channel 0..6
    const int br  = blockIdx.y;   // branch 0..4
    const int tid = threadIdx.x;
    const int N   = NBATCH * LC;  // 102400
    const float* base = wsy + br * (NBATCH * LC * DM);

    float s = 0.f, s2 = 0.f;
    for (int n = tid; n < N; n += 256) {
        float v = base[n * DM + o];
        s  += v;
        s2 += v * v;
    }
    __shared__ float rs[256], rs2[256];
    rs[tid] = s; rs2[tid] = s2;
    __syncthreads();
    for (int st = 128; st > 0; st >>= 1) {
        if (tid < st) { rs[tid] += rs[tid + st]; rs2[tid] += rs2[tid + st]; }
        __syncthreads();
    }
    if (tid == 0) {
        float mu  = rs[0] / (float)N;
        float var = rs2[0] / (float)N - mu * mu;
        float sc  = bn_g[br * DM + o] * rsqrtf(var + 1e-5f);
        stats[(br * DM + o) * 2 + 0] = sc;
        stats[(br * DM + o) * 2 + 1] = bn_b[br * DM + o] - mu * sc;
    }
}

// ===========================================================================
// K3: apply BN, circular conv(3) over time across the 7 channels, ELU.
// One thread per (b,t); computes all 7 output channels.
// ===========================================================================
__global__ __launch_bounds__(256)
void k3_bn_conv_elu(const float* __restrict__ wsy,
                    const float* __restrict__ stats,
                    const float* __restrict__ cvw,   // (5,7,7,3)
                    const float* __restrict__ cvb,   // (5,7)
                    float* __restrict__ out)         // (5,512,200,7)
{
    const int br  = blockIdx.y;
    const int tid = threadIdx.x;
    __shared__ float s_w[DM * DM * 3];
    __shared__ float s_b[DM], s_sc[DM], s_sh[DM];
    for (int i = tid; i < DM * DM * 3; i += 256) s_w[i] = cvw[br * DM * DM * 3 + i];
    if (tid < DM) {
        s_b[tid]  = cvb[br * DM + tid];
        s_sc[tid] = stats[(br * DM + tid) * 2 + 0];
        s_sh[tid] = stats[(br * DM + tid) * 2 + 1];
    }
    __syncthreads();

    const int idx = blockIdx.x * 256 + tid;   // 0 .. 102399 (exact)
    const int b   = idx / LC;
    const int t   = idx % LC;
    const int tm  = (t + LC - 1) % LC;
    const int tp  = (t + 1) % LC;
    const float* base = wsy + (br * NBATCH + b) * LC * DM;

    float tn[DM][3];
    #pragma unroll
    for (int i = 0; i < DM; ++i) {
        tn[i][0] = base[tm * DM + i] * s_sc[i] + s_sh[i];
        tn[i][1] = base[t  * DM + i] * s_sc[i] + s_sh[i];
        tn[i][2] = base[tp * DM + i] * s_sc[i] + s_sh[i];
    }
    #pragma unroll
    for (int o = 0; o < DM; ++o) {
        float v = s_b[o];
        #pragma unroll
        for (int i = 0; i < DM; ++i) {
            v += s_w[(o * DM + i) * 3 + 0] * tn[i][0];
            v += s_w[(o * DM + i) * 3 + 1] * tn[i][1];
            v += s_w[(o * DM + i) * 3 + 2] * tn[i][2];
        }
        out[((br * NBATCH + b) * LC + t) * DM + o] = (v > 0.f) ? v : expm1f(v);
    }
}

// ===========================================================================
extern "C" void kernel_launch(void* const* d_in, const int* in_sizes, int n_in,
                              void* d_out, int out_size, void* d_ws, size_t ws_size,
                              hipStream_t stream)
{
    const float* x    = (const float*)d_in[0];
    const float* ipw  = (const float*)d_in[1];
    const float* cw   = (const float*)d_in[2];
    const float* cb   = (const float*)d_in[3];
    const float* xpw  = (const float*)d_in[4];
    const float* dtw  = (const float*)d_in[5];
    const float* dtb  = (const float*)d_in[6];
    const float* alog = (const float*)d_in[7];
    const float* dpw  = (const float*)d_in[8];
    const float* opw  = (const float*)d_in[9];
    const float* bng  = (const float*)d_in[10];
    const float* bnb  = (const float*)d_in[11];
    const float* cvw  = (const float*)d_in[12];
    const float* cvb  = (const float*)d_in[13];
    float* out = (float*)d_out;

    float* wsy   = (float*)d_ws;                       // 5*512*200*7 floats
    float* stats = wsy + NB * NBATCH * LC * DM;        // 70 floats

    dim3 g1(NBATCH, NB);
    k1_mamba_branch<<<g1, 160, 0, stream>>>(x, ipw, cw, cb, xpw, dtw, dtb,
                                            alog, dpw, opw, wsy);
    dim3 g2(DM, NB);
    k2_bn_stats<<<g2, 256, 0, stream>>>(wsy, bng, bnb, stats);

    dim3 g3((NBATCH * LC) / 256, NB);
    k3_bn_conv_elu<<<g3, 256, 0, stream>>>(wsy, stats, cvw, cvb, out);
}